// CertificateModuleWithDerivatives_16621523435657
// MI455X (gfx1250) — compile-verified
//
#include <hip/hip_runtime.h>
#include <math.h>

// CDNA5 (gfx1250) wave32 WMMA types
typedef __attribute__((ext_vector_type(2))) float v2f;   // 16x4 (or 4x16) f32 tile fragment: 2 VGPRs
typedef __attribute__((ext_vector_type(8))) float v8f;   // 16x16 f32 C/D tile: 8 VGPRs

static __device__ __forceinline__ v8f wmma_f32_16x16x4(v2f a, v2f b, v8f c) {
  // 8-arg pattern: (neg_a, A, neg_b, B, c_mod, C, reuse_a, reuse_b)
  return __builtin_amdgcn_wmma_f32_16x16x4_f32(false, a, false, b, (short)0, c, false, false);
}

#define H 32
#define NIN 2

// One wave (32 lanes) per block; each wave handles 32 samples.
// lane == sample-within-tile for scalar phases; lane%16 == tile column,
// lane/16 == half-wave row split for WMMA fragments (ISA 7.12.2 layouts).
__global__ __launch_bounds__(32) void cert_mlp_deriv_kernel(
    const float* __restrict__ x,   // (B,2)
    const float* __restrict__ W1,  // (32,2)
    const float* __restrict__ b1,  // (32)
    const float* __restrict__ W2,  // (32,32)
    const float* __restrict__ b2,  // (32)
    const float* __restrict__ W3,  // (1,32)
    const float* __restrict__ b3,  // (1)
    float* __restrict__ outbuf,    // out(B) | jac(B*2) | hess(B*4), concatenated
    int Bn)
{
  // ---- block-shared weights (padded against bank conflicts) ----
  __shared__ float sW1[H][2];
  __shared__ float sb1[H];
  __shared__ float sW2[H][33];   // pad 33: fragment-build reads hit distinct banks
  __shared__ float sb2[H];
  __shared__ float sW3[H];
  __shared__ float sW32[H];      // W3 @ W2
  __shared__ float sb3;
  // ---- wave scratch: LDS is the layout shuffle between scalar & WMMA views ----
  __shared__ float sT[H][44];    // t1 then t2, [h][sample]; stride 44: 2-row & 8-row splits conflict-free
  __shared__ float sU[H][76];    // U=ds0*W1 then B1, [h][2*sample+k]; stride 76: same property

  const int lane = threadIdx.x;          // 0..31
  const int col  = lane & 15;            // WMMA tile column
  const int half = lane >> 4;            // half-wave (row-split) index

  // ---- stage weights ----
  sW1[lane][0] = W1[2 * lane];
  sW1[lane][1] = W1[2 * lane + 1];
  sb1[lane] = b1[lane];
  sb2[lane] = b2[lane];
  sW3[lane] = W3[lane];
  if (lane == 0) sb3 = b3[0];
  #pragma unroll
  for (int i = 0; i < H; ++i) sW2[i][lane] = W2[i * H + lane];
  __syncthreads();                       // single-wave WG: lowers to S_NOP

  // W32[h] = sum_i W3[i] * W2[i][h]
  {
    float acc = 0.f;
    #pragma unroll
    for (int i = 0; i < H; ++i) acc = fmaf(sW3[i], sW2[i][lane], acc);
    sW32[lane] = acc;
  }
  __syncthreads();

  // ---- resident W2 A-fragments: tile m rows m*16..m*16+15, K chunk kk (4 wide).
  // A layout (16x4 f32): lane row = col, VGPR0={K+0|K+2}, VGPR1={K+1|K+3} per half-wave.
  v2f wfrag[2][8];
  #pragma unroll
  for (int m = 0; m < 2; ++m)
    #pragma unroll
    for (int kk = 0; kk < 8; ++kk) {
      const int r  = m * 16 + col;
      const int kb = kk * 4 + 2 * half;
      wfrag[m][kk].x = sW2[r][kb];
      wfrag[m][kk].y = sW2[r][kb + 1];
    }

  // ---- sample indices (clamp loads; predicate only final stores: EXEC all-1s at WMMAs) ----
  const int  sraw  = blockIdx.x * 32 + lane;
  const bool valid = (sraw < Bn);
  const int  s     = valid ? sraw : (Bn - 1);

  // ================= P1: layer 1 (scalar, lane = sample) =================
  const float x0 = x[2 * s], x1 = x[2 * s + 1];
  float h00 = 0.f, h01 = 0.f, h11 = 0.f;      // term1 accum: sum_h W32*d2s0*W1kW1l
  #pragma unroll 4
  for (int h = 0; h < H; ++h) {
    const float w0 = sW1[h][0], w1 = sW1[h][1];
    const float z  = fmaf(w0, x0, fmaf(w1, x1, sb1[h]));
    const float t  = tanhf(z);
    const float d  = 1.f - t * t;             // ds0
    const float dd = -2.f * t * d;            // d2s0
    sT[h][lane]        = t;                   // T1[h][n]
    sU[h][2 * lane]    = d * w0;              // U[h][2n+0]
    sU[h][2 * lane + 1] = d * w1;             // U[h][2n+1]
    const float cs = sW32[h] * dd;
    h00 = fmaf(cs, w0 * w0, h00);
    h01 = fmaf(cs, w0 * w1, h01);
    h11 = fmaf(cs, w1 * w1, h11);
  }
  __syncthreads();

  // ================= P2: Z2 = W2 @ T1 + b2 via WMMA; t2 = tanh =================
  // B layout (4x16 f32): VGPR0 = K rows {0|2}, VGPR1 = {1|3} across lane halves.
  v2f tfrag[2][8];
  #pragma unroll
  for (int nh = 0; nh < 2; ++nh)
    #pragma unroll
    for (int kk = 0; kk < 8; ++kk) {
      const int r0 = kk * 4 + 2 * half;
      const int c  = nh * 16 + col;
      tfrag[nh][kk].x = sT[r0][c];
      tfrag[nh][kk].y = sT[r0 + 1][c];
    }
  #pragma unroll
  for (int m = 0; m < 2; ++m)
    #pragma unroll
    for (int nh = 0; nh < 2; ++nh) {
      v8f acc;
      #pragma unroll
      for (int r = 0; r < 8; ++r) acc[r] = sb2[m * 16 + r + 8 * half];  // bias as C (D layout)
      #pragma unroll
      for (int kk = 0; kk < 8; ++kk)
        acc = wmma_f32_16x16x4(wfrag[m][kk], tfrag[nh][kk], acc);
      const int c     = nh * 16 + col;
      const int hbase = m * 16 + 8 * half;
      #pragma unroll
      for (int r = 0; r < 8; ++r)
        sT[hbase + r][c] = tanhf(acc[r]);     // overwrite T1 with T2 (frags already in regs)
    }
  __syncthreads();

  // ================= P4: layer 3 + softplus (scalar) =================
  float z3 = sb3;
  #pragma unroll 4
  for (int h = 0; h < H; ++h) z3 = fmaf(sW3[h], sT[h][lane], z3);
  const float sg   = 1.f / (1.f + __expf(-z3));     // sigmoid = ds2
  const float ds2  = sg;
  const float d2s2 = sg * (1.f - sg);
  const float outv = fmaxf(z3, 0.f) + log1pf(__expf(-fabsf(z3)));  // softplus, overflow-safe

  // ================= P5/P6: B1 = W2 @ U via WMMA (reuses W2 A-frags) ============
  // U columns c = 2*sample + k; 4 column-half tiles; results overwrite sU in place.
  #pragma unroll
  for (int ch = 0; ch < 4; ++ch) {
    const int cb = ch * 16 + col;
    v2f ufrag[8];
    #pragma unroll
    for (int kk = 0; kk < 8; ++kk) {
      const int r0 = kk * 4 + 2 * half;
      ufrag[kk].x = sU[r0][cb];
      ufrag[kk].y = sU[r0 + 1][cb];
    }
    v8f acc0 = {0.f, 0.f, 0.f, 0.f, 0.f, 0.f, 0.f, 0.f};
    v8f acc1 = {0.f, 0.f, 0.f, 0.f, 0.f, 0.f, 0.f, 0.f};
    #pragma unroll
    for (int kk = 0; kk < 8; ++kk) acc0 = wmma_f32_16x16x4(wfrag[0][kk], ufrag[kk], acc0);
    #pragma unroll
    for (int kk = 0; kk < 8; ++kk) acc1 = wmma_f32_16x16x4(wfrag[1][kk], ufrag[kk], acc1);
    __syncthreads();                           // fence U reads before in-place overwrite
    const int hb = 8 * half;
    #pragma unroll
    for (int r = 0; r < 8; ++r) sU[hb + r][cb]      = acc0[r];   // B1 rows m=0
    #pragma unroll
    for (int r = 0; r < 8; ++r) sU[16 + hb + r][cb] = acc1[r];   // B1 rows m=1
  }
  __syncthreads();

  // ================= P7: B2 / jacobian / hessian (scalar, lane = sample) ========
  float B20 = 0.f, B21 = 0.f, H200 = 0.f, H201 = 0.f, H211 = 0.f;
  #pragma unroll 4
  for (int h = 0; h < H; ++h) {
    const float t2  = sT[h][lane];
    const float ds1 = 1.f - t2 * t2;
    const float w3  = sW3[h];
    const float g   = w3 * ds1;                       // W3*ds1 -> B2 weight
    const float b10 = sU[h][2 * lane];
    const float b11 = sU[h][2 * lane + 1];
    B20 = fmaf(g, b10, B20);
    B21 = fmaf(g, b11, B21);
    const float c2 = ds2 * w3 * (-2.f * t2 * ds1);    // ds2*W3*d2s1
    H200 = fmaf(c2, b10 * b10, H200);
    H201 = fmaf(c2, b10 * b11, H201);
    H211 = fmaf(c2, b11 * b11, H211);
  }
  const float j0 = ds2 * B20, j1 = ds2 * B21;
  const float g2 = ds2 * ds2;
  const float Hs00 = fmaf(g2, h00, H200) + d2s2 * B20 * B20;
  const float Hs01 = fmaf(g2, h01, H201) + d2s2 * B20 * B21;
  const float Hs11 = fmaf(g2, h11, H211) + d2s2 * B21 * B21;

  if (valid) {
    outbuf[s] = outv;                                  // out (B,1)
    float* jac = outbuf + (size_t)Bn;                  // jacobian (B,1,2)
    jac[2 * s]     = j0;
    jac[2 * s + 1] = j1;
    float* hes = outbuf + 3 * (size_t)Bn;              // hessian (B,2,2)
    hes[4 * s]     = Hs00;
    hes[4 * s + 1] = Hs01;
    hes[4 * s + 2] = Hs01;
    hes[4 * s + 3] = Hs11;
  }
}

extern "C" void kernel_launch(void* const* d_in, const int* in_sizes, int n_in,
                              void* d_out, int out_size, void* d_ws, size_t ws_size,
                              hipStream_t stream) {
  const float* x  = (const float*)d_in[0];
  const float* W1 = (const float*)d_in[1];
  const float* b1 = (const float*)d_in[2];
  const float* W2 = (const float*)d_in[3];
  const float* b2 = (const float*)d_in[4];
  const float* W3 = (const float*)d_in[5];
  const float* b3 = (const float*)d_in[6];
  const int Bn = in_sizes[0] / 2;                      // x is (B, 2)
  const int blocks = (Bn + 31) / 32;                   // 32 samples per wave/block
  cert_mlp_deriv_kernel<<<blocks, 32, 0, stream>>>(x, W1, b1, W2, b2, W3, b3,
                                                   (float*)d_out, Bn);
}